// NodeConvolution_30657476559413
// MI455X (gfx1250) — compile-verified
//
#include <hip/hip_runtime.h>
#include <hip/hip_bf16.h>
#include <math.h>

// ---------------------------------------------------------------------------
// Hypergraph convolution, algebraically reordered:
//   gathered_messages = segsum(conv * X[snd]) @ Wm^T + segsum(conv) (x) bm
//   gathered_scaling  = segsum(conv2 * H[snd]) @ Ws^T + segsum(conv2) (x) bs
//   out = tanh(gathered_scaling * gathered_messages)
// Phase 1: per-node segment reduction (receivers are sorted -> binary search,
//          no atomics, deterministic).
// Phase 2: dense fp32 WMMA GEMM (V_WMMA_F32_16X16X4_F32) + fused epilogue.
// ---------------------------------------------------------------------------

typedef __attribute__((ext_vector_type(2))) float v2f;
typedef __attribute__((ext_vector_type(8))) float v8f;

#define F_IN  128
#define F_H   64
#define F_OUT 128

__device__ __forceinline__ int lower_bound_i(const int* __restrict__ a, int n, int v) {
    int lo = 0, hi = n;
    while (lo < hi) {
        int mid = (lo + hi) >> 1;
        if (a[mid] < v) lo = mid + 1; else hi = mid;
    }
    return lo;
}

// One wave per destination node; 128 features -> 4 floats/lane.
__global__ __launch_bounds__(256) void seg_reduce_f128(
    const float* __restrict__ feat, const int* __restrict__ snd,
    const int* __restrict__ rcv, const float* __restrict__ conv,
    float* __restrict__ A, float* __restrict__ c, int nSeg, int E)
{
    int wave = (blockIdx.x * blockDim.x + threadIdx.x) >> 5;
    int lane = threadIdx.x & 31;
    if (wave >= nSeg) return;
    int lo = lower_bound_i(rcv, E, wave);
    int hi = lower_bound_i(rcv, E, wave + 1);
    float4 acc = make_float4(0.f, 0.f, 0.f, 0.f);
    float wsum = 0.f;
    for (int e = lo; e < hi; ++e) {
        int s = snd[e];
        float w = conv[e];
        const float4 f = *(const float4*)(feat + (size_t)s * F_IN + lane * 4);
        acc.x += w * f.x; acc.y += w * f.y; acc.z += w * f.z; acc.w += w * f.w;
        wsum += w;
    }
    *(float4*)(A + (size_t)wave * F_IN + lane * 4) = acc;
    if (lane == 0) c[wave] = wsum;
}

// One wave per destination node; 64 features -> 2 floats/lane.
__global__ __launch_bounds__(256) void seg_reduce_f64(
    const float* __restrict__ feat, const int* __restrict__ snd,
    const int* __restrict__ rcv, const float* __restrict__ conv,
    float* __restrict__ A, float* __restrict__ c, int nSeg, int E)
{
    int wave = (blockIdx.x * blockDim.x + threadIdx.x) >> 5;
    int lane = threadIdx.x & 31;
    if (wave >= nSeg) return;
    int lo = lower_bound_i(rcv, E, wave);
    int hi = lower_bound_i(rcv, E, wave + 1);
    float2 acc = make_float2(0.f, 0.f);
    float wsum = 0.f;
    for (int e = lo; e < hi; ++e) {
        int s = snd[e];
        float w = conv[e];
        const float2 f = *(const float2*)(feat + (size_t)s * F_H + lane * 2);
        acc.x += w * f.x; acc.y += w * f.y;
        wsum += w;
    }
    *(float2*)(A + (size_t)wave * F_H + lane * 2) = acc;
    if (lane == 0) c[wave] = wsum;
}

// Block = 256 threads = 8 waves. Block handles one 16-node tile; each wave
// owns a 16-wide slice of the 128 output features. fp32 WMMA, K stepped by 4.
__global__ __launch_bounds__(256) void hyper_gemm_wmma(
    const float* __restrict__ A1, const float* __restrict__ A2,
    const float* __restrict__ c1, const float* __restrict__ c2,
    const float* __restrict__ Wm, const float* __restrict__ bm,
    const float* __restrict__ Ws, const float* __restrict__ bs,
    float* __restrict__ out, int nNodes)
{
    __shared__ float A1s[16][F_IN];
    __shared__ float A2s[16][F_H];
    __shared__ float c1s[16];
    __shared__ float c2s[16];

    const int base = blockIdx.x * 16;
    const int t = threadIdx.x;

    // Cooperative stage of A-tiles into LDS (float4, fully coalesced).
    for (int i = t; i < 16 * F_IN / 4; i += 256) {          // 512 float4
        int row = i >> 5, col = i & 31;
        float4 v = make_float4(0.f, 0.f, 0.f, 0.f);
        if (base + row < nNodes)
            v = *((const float4*)(A1 + (size_t)(base + row) * F_IN) + col);
        ((float4*)&A1s[0][0])[i] = v;
    }
    {
        int i = t;                                           // 256 float4
        int row = i >> 4, col = i & 15;
        float4 v = make_float4(0.f, 0.f, 0.f, 0.f);
        if (base + row < nNodes)
            v = *((const float4*)(A2 + (size_t)(base + row) * F_H) + col);
        ((float4*)&A2s[0][0])[i] = v;
    }
    if (t < 16) {
        c1s[t] = (base + t < nNodes) ? c1[base + t] : 0.f;
        c2s[t] = (base + t < nNodes) ? c2[base + t] : 0.f;
    }
    __syncthreads();

    const int wave = t >> 5;          // 0..7 -> output-feature tile
    const int lane = t & 31;
    const int nbase = wave * 16;
    const int kh = lane >> 4;         // half-wave selects K sub-pair
    const int mn = lane & 15;         // A-row (m) / B-col (n)

    v8f cm = {};   // messages accumulator  (16x16 f32 tile)
    v8f cs = {};   // scaling accumulator

    // messages: A1[16x128] @ Wm^T slice, K=128 in steps of 4
#pragma unroll
    for (int kb = 0; kb < F_IN / 4; ++kb) {
        const int k = kb * 4 + 2 * kh;
        v2f a; a.x = A1s[mn][k]; a.y = A1s[mn][k + 1];
        const float2 wv = *(const float2*)(Wm + (size_t)(nbase + mn) * F_IN + k);
        v2f b; b.x = wv.x; b.y = wv.y;
        cm = __builtin_amdgcn_wmma_f32_16x16x4_f32(
            false, a, false, b, (short)0, cm, false, false);
    }

    // scaling: A2[16x64] @ Ws^T slice, K=64 in steps of 4
#pragma unroll
    for (int kb = 0; kb < F_H / 4; ++kb) {
        const int k = kb * 4 + 2 * kh;
        v2f a; a.x = A2s[mn][k]; a.y = A2s[mn][k + 1];
        const float2 wv = *(const float2*)(Ws + (size_t)(nbase + mn) * F_H + k);
        v2f b; b.x = wv.x; b.y = wv.y;
        cs = __builtin_amdgcn_wmma_f32_16x16x4_f32(
            false, a, false, b, (short)0, cs, false, false);
    }

    // Fused epilogue: bias terms, multiply pipelines, tanh, store.
    const float bmv = bm[nbase + mn];
    const float bsv = bs[nbase + mn];
#pragma unroll
    for (int v = 0; v < 8; ++v) {
        const int M = v + 8 * kh;          // C/D layout: VGPR v, half-wave kh
        const int row = base + M;
        if (row < nNodes) {
            const float x = cm[v] + c1s[M] * bmv;
            const float y = cs[v] + c2s[M] * bsv;
            out[(size_t)row * F_OUT + nbase + mn] = tanhf(y * x);
        }
    }
}

extern "C" void kernel_launch(void* const* d_in, const int* in_sizes, int n_in,
                              void* d_out, int out_size, void* d_ws, size_t ws_size,
                              hipStream_t stream) {
    const float* node_features  = (const float*)d_in[0];
    const float* hedge_features = (const float*)d_in[1];
    const int*   node_snd       = (const int*)d_in[2];
    const int*   node_rcv       = (const int*)d_in[3];
    const float* node_conv      = (const float*)d_in[4];
    const int*   h_snd          = (const int*)d_in[5];
    const int*   h_rcv          = (const int*)d_in[6];
    const float* h_conv         = (const float*)d_in[7];
    const float* Wm             = (const float*)d_in[8];
    const float* bm             = (const float*)d_in[9];
    const float* Ws             = (const float*)d_in[10];
    const float* bs             = (const float*)d_in[11];

    const int N = in_sizes[0] / F_IN;     // 50000
    const int E = in_sizes[2];            // 800000

    float* A1 = (float*)d_ws;                       // [N,128]
    float* A2 = A1 + (size_t)N * F_IN;              // [N,64]
    float* c1 = A2 + (size_t)N * F_H;               // [N]
    float* c2 = c1 + N;                             // [N]

    const int segBlocks = (N + 7) / 8;              // 8 waves/block, 1 wave/node
    seg_reduce_f128<<<segBlocks, 256, 0, stream>>>(
        node_features, node_snd, node_rcv, node_conv, A1, c1, N, E);
    seg_reduce_f64<<<segBlocks, 256, 0, stream>>>(
        hedge_features, h_snd, h_rcv, h_conv, A2, c2, N, E);

    const int tiles = (N + 15) / 16;                // 3125 tiles
    hyper_gemm_wmma<<<tiles, 256, 0, stream>>>(
        A1, A2, c1, c2, Wm, bm, Ws, bs, (float*)d_out, N);
}